// EuclideanDistanceAddressing_81003083203007
// MI455X (gfx1250) — compile-verified
//
#include <hip/hip_runtime.h>
#include <hip/hip_bf16.h>
#include <math.h>

typedef __attribute__((ext_vector_type(2))) float v2f;
typedef __attribute__((ext_vector_type(8))) float v8f;

#define NB 32
#define NM 8192
#define NI 256

// ---------------------------------------------------------------------------
// Kernel 1: projected_key = tanh(key @ W^T + b) via V_WMMA_F32_16X16X4_F32.
// Grid: 32 blocks (2 row-tiles x 16 col-tiles), 1 wave (32 threads) each.
// A-layout (ISA 7.12.2, 32-bit A 16x4): lanes 0-15 hold K=0,1; lanes 16-31 K=2,3
// for the same M rows. B assumed to mirror A (K split across wave halves).
// ---------------------------------------------------------------------------
__global__ __launch_bounds__(32)
void proj_wmma_kernel(const float* __restrict__ key,
                      const float* __restrict__ W,
                      const float* __restrict__ bias,
                      float* __restrict__ pk) {
    const int lane = threadIdx.x;
    const int half = lane >> 4;        // 0: K={0,1}, 1: K={2,3}
    const int lr   = lane & 15;
    const int kofs = half * 2;
    const int r0 = (blockIdx.x >> 4) * 16;   // key-row tile
    const int o0 = (blockIdx.x & 15) * 16;   // output-column tile

    v8f c = {};
    const float* arow = key + (r0 + lr) * NI + kofs;  // A: key rows
    const float* brow = W   + (o0 + lr) * NI + kofs;  // B[k][n] = W[o0+n][i0+k]
#pragma unroll 4
    for (int i0 = 0; i0 < NI; i0 += 4) {
        v2f a  = *(const v2f*)(arow + i0);
        v2f bb = *(const v2f*)(brow + i0);
        c = __builtin_amdgcn_wmma_f32_16x16x4_f32(false, a, false, bb,
                                                  (short)0, c, false, false);
    }
    // D layout: c[v], lanes 0-15 -> M=v, lanes 16-31 -> M=v+8; N = lr
    const float bv = bias[o0 + lr];
#pragma unroll
    for (int v = 0; v < 8; ++v) {
        const int row = r0 + v + half * 8;
        pk[row * NI + o0 + lr] = tanhf(c[v] + bv);
    }
}

// ---------------------------------------------------------------------------
// Kernel 2: pk_norm[b] = sum_i pk[b,i]^2   (one wave per batch)
// ---------------------------------------------------------------------------
__global__ __launch_bounds__(32)
void pknorm_kernel(const float* __restrict__ pk, float* __restrict__ pk_norm) {
    const int b = blockIdx.x;
    const int lane = threadIdx.x;
    float s = 0.f;
#pragma unroll
    for (int j = 0; j < NI / 32; ++j) {
        const float v = pk[b * NI + lane + j * 32];
        s += v * v;
    }
#pragma unroll
    for (int off = 16; off >= 1; off >>= 1)
        s += __shfl_xor(s, off, 32);
    if (lane == 0) pk_norm[b] = s;
}

// ---------------------------------------------------------------------------
// Kernel 3: similarity[b,m] = -( ||pk||^2 - 2 mem.pk + ||mem||^2 )
// One wave per 16-row tile of memory; dots via accumulating WMMA with pk
// broadcast into all 16 B-columns (every D column identical -> robust
// extraction); row norms accumulated in VALU from the same loaded data.
// Grid: B * (M/16/8) = 2048 blocks, 256 threads (8 waves).
// ---------------------------------------------------------------------------
__global__ __launch_bounds__(256)
void distance_wmma_kernel(const float* __restrict__ memory,
                          const float* __restrict__ pk,
                          const float* __restrict__ pk_norm,
                          float* __restrict__ sim_out) {
    __shared__ __align__(16) float s_pk[NI];
    __shared__ float s_dot[8][16];

    const int tid  = threadIdx.x;
    const int wave = tid >> 5;
    const int lane = tid & 31;
    const int half = lane >> 4;
    const int lr   = lane & 15;
    const int kofs = half * 2;

    const int b     = blockIdx.x >> 6;                 // 64 blocks per batch
    const int mtile = (blockIdx.x & 63) * 8 + wave;    // 512 tiles per batch
    const int row   = mtile * 16 + lr;                 // A rows: M = lr (both halves)

    s_pk[tid] = pk[b * NI + tid];
    __syncthreads();

    const float* mrow = memory + ((size_t)b * NM + row) * NI + kofs;
    v8f c = {};
    float sq = 0.f;
#pragma unroll 4
    for (int i0 = 0; i0 < NI; i0 += 4) {
        v2f a  = *(const v2f*)(mrow + i0);             // A: 16 memory rows x 4
        v2f bb = *(const v2f*)(&s_pk[i0 + kofs]);      // B: pk chunk, all columns
        sq += a.x * a.x + a.y * a.y;
        c = __builtin_amdgcn_wmma_f32_16x16x4_f32(false, a, false, bb,
                                                  (short)0, c, false, false);
    }
    // full row norm: combine the two K-halves (lane l <-> l^16, same row lr)
    const float norm = sq + __shfl_xor(sq, 16, 32);

    // stash dots (identical across the 16 lanes of each half)
#pragma unroll
    for (int v = 0; v < 8; ++v)
        s_dot[wave][v + half * 8] = c[v];
    __syncthreads();

    if (half == 0) {
        const float dist = pk_norm[b] + norm - 2.f * s_dot[wave][lr];
        sim_out[(size_t)b * NM + mtile * 16 + lr] = -dist;
    }
}

// ---------------------------------------------------------------------------
// Kernel 4: per-batch softmax of beta*sim -> weights. One block per batch.
// ---------------------------------------------------------------------------
__global__ __launch_bounds__(256)
void softmax_kernel(const float* __restrict__ sim,
                    const float* __restrict__ beta,
                    float* __restrict__ weights) {
    __shared__ float red[256];
    const int b = blockIdx.x, tid = threadIdx.x;
    const float bt = beta[b];
    const float* s = sim + (size_t)b * NM;
    float* w = weights + (size_t)b * NM;

    float mx = -INFINITY;
    for (int m = tid; m < NM; m += 256) mx = fmaxf(mx, bt * s[m]);
    red[tid] = mx; __syncthreads();
    for (int st = 128; st > 0; st >>= 1) {
        if (tid < st) red[tid] = fmaxf(red[tid], red[tid + st]);
        __syncthreads();
    }
    const float gmx = red[0]; __syncthreads();

    float sum = 0.f;
    for (int m = tid; m < NM; m += 256) {
        const float e = expf(bt * s[m] - gmx);
        w[m] = e; sum += e;
    }
    red[tid] = sum; __syncthreads();
    for (int st = 128; st > 0; st >>= 1) {
        if (tid < st) red[tid] += red[tid + st];
        __syncthreads();
    }
    const float inv = 1.f / red[0]; __syncthreads();
    for (int m = tid; m < NM; m += 256) w[m] *= inv;
}

// ---------------------------------------------------------------------------
// Kernel 5: partial read_vector. Grid: B*32 blocks (chunks of 256 rows),
// 256 threads (one column each, coalesced streaming of memory).
// ---------------------------------------------------------------------------
__global__ __launch_bounds__(256)
void readvec_partial_kernel(const float* __restrict__ memory,
                            const float* __restrict__ weights,
                            float* __restrict__ partial) {
    __shared__ float sw[256];
    const int b  = blockIdx.x >> 5;
    const int ch = blockIdx.x & 31;
    const int tid = threadIdx.x;

    sw[tid] = weights[(size_t)b * NM + ch * 256 + tid];
    __syncthreads();

    const float* base = memory + ((size_t)b * NM + ch * 256) * NI + tid;
    float acc = 0.f;
#pragma unroll 8
    for (int j = 0; j < 256; ++j)
        acc += sw[j] * base[(size_t)j * NI];
    partial[((size_t)b * 32 + ch) * NI + tid] = acc;
}

// ---------------------------------------------------------------------------
// Kernel 6: reduce partials -> read_vector. 32 blocks x 256 threads.
// ---------------------------------------------------------------------------
__global__ __launch_bounds__(256)
void readvec_reduce_kernel(const float* __restrict__ partial,
                           float* __restrict__ rv) {
    const int idx = blockIdx.x * 256 + threadIdx.x;   // 0 .. B*I-1
    const int b = idx >> 8, i = idx & 255;
    float s = 0.f;
#pragma unroll
    for (int ch = 0; ch < 32; ++ch)
        s += partial[((size_t)b * 32 + ch) * NI + i];
    rv[idx] = s;
}

// ---------------------------------------------------------------------------
extern "C" void kernel_launch(void* const* d_in, const int* in_sizes, int n_in,
                              void* d_out, int out_size, void* d_ws, size_t ws_size,
                              hipStream_t stream) {
    const float* memory = (const float*)d_in[0];   // [B, M, I]
    const float* key    = (const float*)d_in[1];   // [B, I]
    const float* beta   = (const float*)d_in[2];   // [B]
    const float* W      = (const float*)d_in[3];   // [I, I]
    const float* bias   = (const float*)d_in[4];   // [I]

    float* out = (float*)d_out;
    float* weights = out;                          // [B, M]
    float* rv      = out + (size_t)NB * NM;        // [B, I]
    float* sim     = rv  + (size_t)NB * NI;        // [B, M]

    float* ws      = (float*)d_ws;
    float* pk      = ws;                           // [B, I]      8192
    float* pk_norm = ws + NB * NI;                 // [B]         32
    float* partial = ws + NB * NI + 256;           // [B, 32, I]  262144

    proj_wmma_kernel<<<32, 32, 0, stream>>>(key, W, bias, pk);
    pknorm_kernel<<<NB, 32, 0, stream>>>(pk, pk_norm);
    distance_wmma_kernel<<<NB * 64, 256, 0, stream>>>(memory, pk, pk_norm, sim);
    softmax_kernel<<<NB, 256, 0, stream>>>(sim, beta, weights);
    readvec_partial_kernel<<<NB * 32, 256, 0, stream>>>(memory, weights, partial);
    readvec_reduce_kernel<<<(NB * NI) / 256, 256, 0, stream>>>(partial, rv);
}